// BaseGNN_43473658970342
// MI455X (gfx1250) — compile-verified
//
#include <hip/hip_runtime.h>

// ---------------------------------------------------------------------------
// GCN on gfx1250 (MI455X, wave32, WMMA + async-to-LDS).
// Reassociation: S@(h@W) == (S@h)@W  -> aggregate in the *small* dim first.
// pool(S@h2@W3+b3) == pool(S@h2)@W3+b3 -> final GEMM is only 64x128x256.
// Dense math uses V_WMMA_F32_16X16X4_F32 (f32 in/acc, matches f32 reference).
// GEMM A-tiles staged with GLOBAL_LOAD_ASYNC_TO_LDS_B128 (ASYNCcnt path).
// ---------------------------------------------------------------------------

typedef __attribute__((ext_vector_type(2))) float v2f;
typedef __attribute__((ext_vector_type(8))) float v8f;

#define GCN_N   100000
#define GCN_B   64
#define GCN_IND 20
#define GCN_HID 128
#define GCN_OUT 256
#define GCN_EPS 1e-5f

// ---------------- simple fills / degree / rsqrt ----------------------------

__global__ void k_fill(float* __restrict__ p, float v, long long n) {
    long long i = (long long)blockIdx.x * blockDim.x + threadIdx.x;
    if (i < n) p[i] = v;
}

__global__ void k_deg(const int* __restrict__ dst, float* __restrict__ deg, int e) {
    int i = blockIdx.x * blockDim.x + threadIdx.x;
    if (i < e) atomicAdd(&deg[dst[i]], 1.0f);
}

__global__ void k_rsqrt(float* __restrict__ p, int n) {
    int i = blockIdx.x * blockDim.x + threadIdx.x;
    if (i < n) p[i] = rsqrtf(p[i]);
}

// ---------------- normalized scatter aggregation ----------------------------
// out[d] += dinv[s]*dinv[d]*in[s].  One block per edge, threadIdx.x = feature:
// src/dst/dinv loads are block-uniform, the row gather is fully coalesced.
// Blocks [0,E) are real edges; blocks [E, E+N) are self loops.

__global__ void k_aggregate(const float* __restrict__ in, float* __restrict__ out,
                            const int* __restrict__ src, const int* __restrict__ dst,
                            const float* __restrict__ dinv, int E, int F) {
    const int e = blockIdx.x;
    const int f = threadIdx.x;
    if (f >= F) return;
    int s, d;
    if (e < E) { s = src[e]; d = dst[e]; }
    else       { s = e - E; d = s; }
    const float nrm = dinv[s] * dinv[d];
    atomicAdd(&out[(long long)d * F + f], nrm * in[(long long)s * F + f]);
}

// ---------------- WMMA f32 GEMM: C[M x Nc] = A[M x K] @ W[K x Nc] + bias ----
// One block (8 waves) per 16-row M-tile; wave w covers N-tiles w, w+8, ...
// A tile staged straight into LDS via GLOBAL_LOAD_ASYNC_TO_LDS_B128; padded
// row stride of 132 floats (528 B = 33*16) keeps every chunk 16B-aligned in
// both LDS and global while dodging bank conflicts.
// WMMA layouts per CDNA5 ISA 7.12.2:
//   A 16x4 : lanes 0-15 hold K={k0,k0+1}, lanes 16-31 hold K={k0+2,k0+3}
//   B 4x16 : mirror of A with N striped across lanes
//   C/D    : VGPR r, lanes 0-15 -> M=r, lanes 16-31 -> M=r+8
// Requires M % 16 == 0 (100000 and 64 both are), K % 4 == 0, Nc % 16 == 0.

__global__ void __launch_bounds__(256)
k_wmma_gemm_bias(const float* __restrict__ A, const float* __restrict__ W,
                 const float* __restrict__ bias, float* __restrict__ C,
                 int M, int K, int Nc) {
    __shared__ float As[16][132];   // 528 B row stride (multiple of 16 B)
    const int m0  = blockIdx.x * 16;
    const int tid = threadIdx.x;

    // async-stage the 16 x K A-tile: one B128 copy per 16-byte chunk
    {
        const unsigned lds_base = (unsigned)(unsigned long long)(void*)&As[0][0];
        const int cpr = K >> 2;                 // 16B chunks per row
        for (int c = tid; c < 16 * cpr; c += 256) {
            const int r  = c / cpr;
            const int ch = c % cpr;
            const float*   g     = A + (long long)(m0 + r) * K + (ch << 2);
            const unsigned l_off = lds_base + (unsigned)(r * 528 + (ch << 4));
            asm volatile("global_load_async_to_lds_b128 %0, %1, off"
                         :: "v"(l_off), "v"(g) : "memory");
        }
        asm volatile("s_wait_asynccnt 0x0" ::: "memory");
    }
    __syncthreads();

    const int lane = tid & 31;
    const int wave = tid >> 5;
    const int half = lane >> 4;   // 0: lanes 0-15, 1: lanes 16-31
    const int l    = lane & 15;
    const int ntiles = Nc >> 4;

    for (int nt = wave; nt < ntiles; nt += 8) {
        const int n0 = nt << 4;
        v8f acc;
        const float bv = bias[n0 + l];
#pragma unroll
        for (int r = 0; r < 8; ++r) acc[r] = bv;

        for (int k0 = 0; k0 < K; k0 += 4) {
            v2f a, b;
            a.x = As[l][k0 + 2 * half];
            a.y = As[l][k0 + 2 * half + 1];
            b.x = W[(long long)(k0 + 2 * half)     * Nc + n0 + l];
            b.y = W[(long long)(k0 + 2 * half + 1) * Nc + n0 + l];
            acc = __builtin_amdgcn_wmma_f32_16x16x4_f32(
                false, a, false, b, (short)0, acc, false, false);
        }
#pragma unroll
        for (int r = 0; r < 8; ++r) {
            int m = m0 + r + 8 * half;
            C[(long long)m * Nc + n0 + l] = acc[r];
        }
    }
}

// ---------------- LayerNorm(conv + res)*g + b, ReLU; one block per row ------
// Safe even when `out` aliases `conv` or `res` (reads precede the single write).

__global__ void __launch_bounds__(128)
k_ln_relu(const float* __restrict__ conv, const float* __restrict__ res,
          const float* __restrict__ g, const float* __restrict__ b,
          float* __restrict__ out) {
    __shared__ float sh[128];
    const int n = blockIdx.x;
    const int f = threadIdx.x;
    const long long base = (long long)n * GCN_HID;
    float h = conv[base + f] + res[base + f];

    sh[f] = h; __syncthreads();
    for (int s = 64; s > 0; s >>= 1) { if (f < s) sh[f] += sh[f + s]; __syncthreads(); }
    float mu = sh[0] * (1.0f / GCN_HID);
    __syncthreads();

    float dv = h - mu;
    sh[f] = dv * dv; __syncthreads();
    for (int s = 64; s > 0; s >>= 1) { if (f < s) sh[f] += sh[f + s]; __syncthreads(); }
    float var = sh[0] * (1.0f / GCN_HID);

    float y = dv * rsqrtf(var + GCN_EPS) * g[f] + b[f];
    out[base + f] = fmaxf(y, 0.0f);
}

// ---------------- per-graph pooling -----------------------------------------

__global__ void __launch_bounds__(128)
k_pool_sum(const float* __restrict__ A, const int* __restrict__ batch,
           float* __restrict__ pooled, float* __restrict__ cnt) {
    const int n = blockIdx.x;
    const int f = threadIdx.x;
    const int gi = batch[n];
    atomicAdd(&pooled[(long long)gi * GCN_HID + f], A[(long long)n * GCN_HID + f]);
    if (f == 0) atomicAdd(&cnt[gi], 1.0f);
}

__global__ void __launch_bounds__(128)
k_pool_div(float* __restrict__ pooled, const float* __restrict__ cnt) {
    const int gi = blockIdx.x;
    const int f  = threadIdx.x;
    pooled[(long long)gi * GCN_HID + f] /= fmaxf(cnt[gi], 1.0f);
}

// ---------------------------------------------------------------------------

extern "C" void kernel_launch(void* const* d_in, const int* in_sizes, int n_in,
                              void* d_out, int out_size, void* d_ws, size_t ws_size,
                              hipStream_t stream) {
    // setup_inputs() order
    const float* x     = (const float*)d_in[0];
    const int*   edge  = (const int*)  d_in[1];   // [2,E]: row0=src, row1=dst
    const int*   batch = (const int*)  d_in[2];
    const float* W1    = (const float*)d_in[3];
    const float* b1    = (const float*)d_in[4];
    const float* W2    = (const float*)d_in[5];
    const float* b2    = (const float*)d_in[6];
    const float* W3    = (const float*)d_in[7];
    const float* b3    = (const float*)d_in[8];
    const float* resW  = (const float*)d_in[9];
    const float* resb  = (const float*)d_in[10];
    const float* ln1g  = (const float*)d_in[11];
    const float* ln1b  = (const float*)d_in[12];
    const float* ln2g  = (const float*)d_in[13];
    const float* ln2b  = (const float*)d_in[14];
    float* out = (float*)d_out;

    const int Nn = GCN_N;
    const int E  = in_sizes[1] / 2;
    const int* src = edge;
    const int* dst = edge + E;

    // workspace layout (floats); every buffer start stays 16B-aligned
    float* ws = (float*)d_ws;
    float* dinv   = ws;                                       // N
    float* aggx   = dinv + Nn;                                // N*20
    float* bufT   = aggx + (size_t)Nn * GCN_IND;              // N*128
    float* bufR   = bufT + (size_t)Nn * GCN_HID;              // N*128
    float* bufA   = bufR + (size_t)Nn * GCN_HID;              // N*128
    float* pooled = bufA + (size_t)Nn * GCN_HID;              // B*128
    float* cnt    = pooled + (size_t)GCN_B * GCN_HID;         // B

    const int ET = E + Nn;   // edges including self loops

    // 1) symmetric GCN normalization: deg (self loops -> init 1), dinv = rsqrt
    k_fill<<<(Nn + 255) / 256, 256, 0, stream>>>(dinv, 1.0f, Nn);
    k_deg<<<(E + 255) / 256, 256, 0, stream>>>(dst, dinv, E);
    k_rsqrt<<<(Nn + 255) / 256, 256, 0, stream>>>(dinv, Nn);

    // 2) layer 1 (aggregate in 20-dim, then GEMM):  conv1 = (S@x)@W1 + b1
    k_fill<<<(unsigned)(((long long)Nn * GCN_IND + 255) / 256), 256, 0, stream>>>(
        aggx, 0.0f, (long long)Nn * GCN_IND);
    k_aggregate<<<ET, 32, 0, stream>>>(x, aggx, src, dst, dinv, E, GCN_IND);
    k_wmma_gemm_bias<<<Nn / 16, 256, 0, stream>>>(aggx, W1, b1, bufT, Nn, GCN_IND, GCN_HID);
    //    residual = x @ res_W + res_b
    k_wmma_gemm_bias<<<Nn / 16, 256, 0, stream>>>(x, resW, resb, bufR, Nn, GCN_IND, GCN_HID);
    //    h1 = relu(LN(conv1 + residual))  (written in place into bufT)
    k_ln_relu<<<Nn, 128, 0, stream>>>(bufT, bufR, ln1g, ln1b, bufT);

    // 3) layer 2:  conv2 = (S@h1)@W2 + b2 ; h2 = relu(LN(conv2 + h1))
    k_fill<<<(unsigned)(((long long)Nn * GCN_HID + 255) / 256), 256, 0, stream>>>(
        bufA, 0.0f, (long long)Nn * GCN_HID);
    k_aggregate<<<ET, GCN_HID, 0, stream>>>(bufT, bufA, src, dst, dinv, E, GCN_HID);
    k_wmma_gemm_bias<<<Nn / 16, 256, 0, stream>>>(bufA, W2, b2, bufR, Nn, GCN_HID, GCN_HID);
    k_ln_relu<<<Nn, 128, 0, stream>>>(bufR, bufT, ln2g, ln2b, bufT);   // h2 -> bufT

    // 4) layer 3 + pool:  out = meanpool(S@h2) @ W3 + b3   (pool commutes with GEMM)
    k_fill<<<(unsigned)(((long long)Nn * GCN_HID + 255) / 256), 256, 0, stream>>>(
        bufA, 0.0f, (long long)Nn * GCN_HID);
    k_aggregate<<<ET, GCN_HID, 0, stream>>>(bufT, bufA, src, dst, dinv, E, GCN_HID);
    k_fill<<<(GCN_B * GCN_HID + 255) / 256, 256, 0, stream>>>(pooled, 0.0f, GCN_B * GCN_HID);
    k_fill<<<1, 64, 0, stream>>>(cnt, 0.0f, GCN_B);
    k_pool_sum<<<Nn, 128, 0, stream>>>(bufA, batch, pooled, cnt);
    k_pool_div<<<GCN_B, 128, 0, stream>>>(pooled, cnt);
    // tiny 64x128x256 WMMA GEMM straight into d_out
    k_wmma_gemm_bias<<<GCN_B / 16, 256, 0, stream>>>(pooled, W3, b3, out, GCN_B, GCN_HID, GCN_OUT);
}